// S4DLayer_21414706938532
// MI455X (gfx1250) — compile-verified
//
#include <hip/hip_runtime.h>
#include <hip/hip_bf16.h>

// ---------------------------------------------------------------------------
// S4D layer on MI455X (gfx1250, wave32).
//   out = (conv_K(x@W_in^T+b_in) + D*u) @ W_out^T + b_out
// FFT conv replaced by equivalent diagonal-SSM chunked scan.
// GEMMs: V_WMMA_F32_16X16X4_F32 (fp32, matches reference precision).
// ---------------------------------------------------------------------------

typedef __attribute__((ext_vector_type(2))) float v2f;
typedef __attribute__((ext_vector_type(8))) float v8f;

constexpr int DM  = 512;     // d_model
constexpr int NST = 64;      // d_state
constexpr int BSZ = 4;
constexpr int TLEN = 4096;
constexpr int TC  = 256;     // scan chunk length
constexpr int NC  = TLEN / TC;   // 16 chunks

// GEMM tiling
constexpr int BM = 128, BN = 64, BK = 64;
constexpr int LDA_PAD = 68;      // BK+4: float4-aligned rows, conflict-free frag reads
constexpr int LDB_PAD = 68;
constexpr int LDT_PAD = 132;     // BM+4 staging stride
constexpr int RED_PAD = 36;      // 32+4: conflict-free transpose-reduce

// workspace layout (floats)
constexpr size_t U_OFF    = 0;                              // (B*DM, T) u
constexpr size_t Y_OFF    = U_OFF + (size_t)BSZ*DM*TLEN;    // (B*DM, T) y
constexpr size_t ABAR_OFF = Y_OFF + (size_t)BSZ*DM*TLEN;    // (DM, NST)
constexpr size_t CB_OFF   = ABAR_OFF + (size_t)DM*NST;      // (DM, NST)
constexpr size_t S_OFF    = CB_OFF + (size_t)DM*NST;        // (B*DM, NC, NST)
constexpr size_t H_OFF    = S_OFF + (size_t)BSZ*DM*NC*NST;  // (B*DM, NC, NST)

__device__ __forceinline__ float lane_bcast(float v, int j) {
    return __builtin_bit_cast(float,
        __builtin_amdgcn_readlane(__builtin_bit_cast(int, v), j));
}

// ---------------- discretization precompute --------------------------------
__global__ __launch_bounds__(256) void k_precompute(
    const float* __restrict__ A_log, const float* __restrict__ Cm,
    const float* __restrict__ log_dt, float* __restrict__ abar,
    float* __restrict__ cbw)
{
    int idx = blockIdx.x * 256 + threadIdx.x;          // 0 .. DM*NST-1
    int d = idx >> 6;
    float A  = -expf(A_log[idx]);
    float dt = expf(log_dt[d]);
    float a  = expf(A * dt);                           // A_bar
    float Bb = (a - 1.0f) / (A * 8.0f);                // (A_bar-1)/(A*sqrt(64))
    abar[idx] = a;
    cbw[idx]  = Cm[idx] * Bb;
}

// ---------------- input projection: u = x @ W_in^T + b_in ------------------
// Writes u transposed into (b, d, t) layout for the scan.
__global__ __launch_bounds__(256) void k_proj_in(
    const float* __restrict__ x, const float* __restrict__ W,
    const float* __restrict__ bias, float* __restrict__ u_out)
{
    __shared__ __align__(16) float As[BM * LDA_PAD];
    __shared__ __align__(16) float Bs[BN * LDB_PAD];

    const int tid  = threadIdx.x;
    const int lane = tid & 31;
    const int wave = tid >> 5;
    const int m0 = blockIdx.x * BM;      // row = b*T + t
    const int e0 = blockIdx.y * BN;      // output feature base

    v8f c[4];
    #pragma unroll
    for (int i = 0; i < 4; ++i) c[i] = (v8f){0,0,0,0,0,0,0,0};

    const int mrow = wave * 16 + (lane & 15);
    const int hi2  = (lane >> 4) << 1;   // K sub-offset 0/2

    for (int kb = 0; kb < DM; kb += BK) {
        __syncthreads();
        #pragma unroll
        for (int r = 0; r < 8; ++r) {                  // A tile 128x64
            int idx = tid + r * 256, row = idx >> 4, c4 = (idx & 15) << 2;
            float4 v = *(const float4*)(x + (size_t)(m0 + row) * DM + kb + c4);
            *(float4*)(As + row * LDA_PAD + c4) = v;
        }
        #pragma unroll
        for (int r = 0; r < 4; ++r) {                  // B tile 64x64 (rows of W_in)
            int idx = tid + r * 256, row = idx >> 4, c4 = (idx & 15) << 2;
            float4 v = *(const float4*)(W + (size_t)(e0 + row) * DM + kb + c4);
            *(float4*)(Bs + row * LDB_PAD + c4) = v;
        }
        __syncthreads();
        for (int ks = 0; ks < BK; ks += 4) {
            v2f a = *(const v2f*)(As + mrow * LDA_PAD + ks + hi2);
            #pragma unroll
            for (int i = 0; i < 4; ++i) {
                int nrow = i * 16 + (lane & 15);
                v2f b = *(const v2f*)(Bs + nrow * LDB_PAD + ks + hi2);
                c[i] = __builtin_amdgcn_wmma_f32_16x16x4_f32(
                    false, a, false, b, (short)0, c[i], false, false);
            }
        }
    }

    // stage transpose through LDS, write (b, e, t) coalesced with +bias
    __syncthreads();
    const int lo = lane & 15;
    const int mbase = wave * 16 + ((lane >> 4) << 3);
    #pragma unroll
    for (int i = 0; i < 4; ++i) {
        int e_loc = i * 16 + lo;
        #pragma unroll
        for (int v = 0; v < 8; ++v)
            As[e_loc * LDT_PAD + mbase + v] = c[i][v];
    }
    __syncthreads();
    const int bidx = m0 >> 12, tbase = m0 & (TLEN - 1);
    #pragma unroll
    for (int r = 0; r < 8; ++r) {
        int idx = tid + r * 256, row = idx >> 5, c4 = (idx & 31) << 2;
        float4 v = *(const float4*)(As + row * LDT_PAD + c4);
        float bb = bias[e0 + row];
        v.x += bb; v.y += bb; v.z += bb; v.w += bb;
        *(float4*)(u_out + ((size_t)(bidx * DM + e0 + row)) * TLEN + tbase + c4) = v;
    }
}

// ---------------- scan pass A: chunk-local end states (h_in = 0) -----------
__global__ __launch_bounds__(256) void k_scan_state(
    const float* __restrict__ u, const float* __restrict__ abar,
    float* __restrict__ s)
{
    const int lane = threadIdx.x & 31;
    const int wid  = (blockIdx.x * 256 + threadIdx.x) >> 5;   // bd*NC + c
    const int c  = wid & (NC - 1);
    const int bd = wid / NC;
    const int d  = bd & (DM - 1);
    float a0 = abar[d * NST + lane], a1 = abar[d * NST + 32 + lane];
    float h0 = 0.f, h1 = 0.f;
    const float* ub = u + (size_t)bd * TLEN + c * TC;
    for (int tb = 0; tb < TC; tb += 32) {
        float uv = ub[tb + lane];
        #pragma unroll
        for (int j = 0; j < 32; ++j) {
            float uj = lane_bcast(uv, j);
            h0 = fmaf(a0, h0, uj);
            h1 = fmaf(a1, h1, uj);
        }
    }
    s[(size_t)wid * NST + lane]      = h0;
    s[(size_t)wid * NST + 32 + lane] = h1;
}

// ---------------- scan pass B: carry states across chunks ------------------
__global__ __launch_bounds__(256) void k_scan_carry(
    const float* __restrict__ s, const float* __restrict__ abar,
    float* __restrict__ hin)
{
    const int lane = threadIdx.x & 31;
    const int bd   = (blockIdx.x * 256 + threadIdx.x) >> 5;   // 0..B*DM-1
    const int d    = bd & (DM - 1);
    float a0 = abar[d * NST + lane], a1 = abar[d * NST + 32 + lane];
    float p0 = a0, p1 = a1;
    #pragma unroll
    for (int i = 0; i < 8; ++i) { p0 *= p0; p1 *= p1; }       // a^256
    float h0 = 0.f, h1 = 0.f;
    for (int c = 0; c < NC; ++c) {
        size_t off = ((size_t)bd * NC + c) * NST;
        hin[off + lane] = h0; hin[off + 32 + lane] = h1;
        h0 = fmaf(p0, h0, s[off + lane]);
        h1 = fmaf(p1, h1, s[off + 32 + lane]);
    }
}

// ---------------- scan pass C: outputs with correct initial state ----------
__global__ __launch_bounds__(256) void k_scan_out(
    const float* __restrict__ u, const float* __restrict__ abar,
    const float* __restrict__ cbw, const float* __restrict__ dskip,
    const float* __restrict__ hin, float* __restrict__ yout)
{
    __shared__ __align__(16) float red[8 * 32 * RED_PAD];
    const int lane  = threadIdx.x & 31;
    const int wslot = (threadIdx.x >> 5) * 32 * RED_PAD;
    const int wid = (blockIdx.x * 256 + threadIdx.x) >> 5;
    const int c  = wid & (NC - 1);
    const int bd = wid / NC;
    const int d  = bd & (DM - 1);
    float a0  = abar[d * NST + lane], a1  = abar[d * NST + 32 + lane];
    float cb0 = cbw[d * NST + lane],  cb1 = cbw[d * NST + 32 + lane];
    float dsk = dskip[d];
    size_t hoff = (size_t)wid * NST;
    float h0 = hin[hoff + lane], h1 = hin[hoff + 32 + lane];
    const float* ub = u + (size_t)bd * TLEN + c * TC;
    float* yb = yout + (size_t)bd * TLEN + c * TC;

    for (int tb = 0; tb < TC; tb += 32) {
        float uv = ub[tb + lane];
        #pragma unroll
        for (int j = 0; j < 32; ++j) {
            float uj = lane_bcast(uv, j);
            h0 = fmaf(a0, h0, uj);
            h1 = fmaf(a1, h1, uj);
            float p = fmaf(cb1, h1, cb0 * h0);
            red[wslot + j * RED_PAD + lane] = p;           // [t][n-lane]
        }
        asm volatile("s_wait_dscnt 0" ::: "memory");       // intra-wave LDS RAW
        const float4* rowp = (const float4*)(red + wslot + lane * RED_PAD);
        float acc = 0.f;
        #pragma unroll
        for (int q = 0; q < 8; ++q) {                      // sum 32 n-partials
            float4 v = rowp[q];
            acc += (v.x + v.y) + (v.z + v.w);
        }
        yb[tb + lane] = fmaf(dsk, uv, acc);                // + D_skip * u
        asm volatile("" ::: "memory");
    }
}

// ---------------- output projection: out = y^T @ W_out^T + b_out -----------
// A-tile loaded with LDS transpose from (b, d, t) layout.
__global__ __launch_bounds__(256) void k_proj_out(
    const float* __restrict__ y, const float* __restrict__ W,
    const float* __restrict__ bias, float* __restrict__ out)
{
    __shared__ __align__(16) float As[BM * LDA_PAD];
    __shared__ __align__(16) float Bs[BN * LDB_PAD];

    const int tid  = threadIdx.x;
    const int lane = tid & 31;
    const int wave = tid >> 5;
    const int m0 = blockIdx.x * BM;
    const int e0 = blockIdx.y * BN;
    const int bidx = m0 >> 12, tbase = m0 & (TLEN - 1);

    v8f c[4];
    #pragma unroll
    for (int i = 0; i < 4; ++i) c[i] = (v8f){0,0,0,0,0,0,0,0};

    const int mrow = wave * 16 + (lane & 15);
    const int hi2  = (lane >> 4) << 1;

    for (int kb = 0; kb < DM; kb += BK) {
        __syncthreads();
        #pragma unroll
        for (int r = 0; r < 8; ++r) {                  // A tile: transpose (d,t)->(t,d)
            int idx = tid + r * 256, drow = idx >> 5, tc4 = (idx & 31) << 2;
            float4 v = *(const float4*)(y + ((size_t)(bidx * DM + kb + drow)) * TLEN
                                          + tbase + tc4);
            As[(tc4 + 0) * LDA_PAD + drow] = v.x;
            As[(tc4 + 1) * LDA_PAD + drow] = v.y;
            As[(tc4 + 2) * LDA_PAD + drow] = v.z;
            As[(tc4 + 3) * LDA_PAD + drow] = v.w;
        }
        #pragma unroll
        for (int r = 0; r < 4; ++r) {                  // B tile: rows of W_out
            int idx = tid + r * 256, row = idx >> 4, c4 = (idx & 15) << 2;
            float4 v = *(const float4*)(W + (size_t)(e0 + row) * DM + kb + c4);
            *(float4*)(Bs + row * LDB_PAD + c4) = v;
        }
        __syncthreads();
        for (int ks = 0; ks < BK; ks += 4) {
            v2f a = *(const v2f*)(As + mrow * LDA_PAD + ks + hi2);
            #pragma unroll
            for (int i = 0; i < 4; ++i) {
                int nrow = i * 16 + (lane & 15);
                v2f b = *(const v2f*)(Bs + nrow * LDB_PAD + ks + hi2);
                c[i] = __builtin_amdgcn_wmma_f32_16x16x4_f32(
                    false, a, false, b, (short)0, c[i], false, false);
            }
        }
    }

    const int lo = lane & 15;
    const int mbase = wave * 16 + ((lane >> 4) << 3);
    #pragma unroll
    for (int i = 0; i < 4; ++i) {
        int e = e0 + i * 16 + lo;
        float bb = bias[e];
        #pragma unroll
        for (int v = 0; v < 8; ++v)
            out[(size_t)(m0 + mbase + v) * DM + e] = c[i][v] + bb;
    }
}

// ---------------------------------------------------------------------------
extern "C" void kernel_launch(void* const* d_in, const int* in_sizes, int n_in,
                              void* d_out, int out_size, void* d_ws, size_t ws_size,
                              hipStream_t stream)
{
    const float* x     = (const float*)d_in[0];
    const float* W_in  = (const float*)d_in[1];
    const float* b_in  = (const float*)d_in[2];
    const float* A_log = (const float*)d_in[3];
    const float* Cm    = (const float*)d_in[4];
    const float* Dsk   = (const float*)d_in[5];
    const float* logdt = (const float*)d_in[6];
    const float* W_out = (const float*)d_in[7];
    const float* b_out = (const float*)d_in[8];
    float* out = (float*)d_out;

    float* ws      = (float*)d_ws;
    float* u_ws    = ws + U_OFF;
    float* y_ws    = ws + Y_OFF;
    float* abar_ws = ws + ABAR_OFF;
    float* cb_ws   = ws + CB_OFF;
    float* s_ws    = ws + S_OFF;
    float* h_ws    = ws + H_OFF;

    k_precompute<<<DM * NST / 256, 256, 0, stream>>>(A_log, Cm, logdt, abar_ws, cb_ws);
    k_proj_in<<<dim3(BSZ * TLEN / BM, DM / BN), 256, 0, stream>>>(x, W_in, b_in, u_ws);
    k_scan_state<<<BSZ * DM * NC / 8, 256, 0, stream>>>(u_ws, abar_ws, s_ws);
    k_scan_carry<<<BSZ * DM / 8, 256, 0, stream>>>(s_ws, abar_ws, h_ws);
    k_scan_out<<<BSZ * DM * NC / 8, 256, 0, stream>>>(u_ws, abar_ws, cb_ws, Dsk, h_ws, y_ws);
    k_proj_out<<<dim3(BSZ * TLEN / BM, DM / BN), 256, 0, stream>>>(y_ws, W_out, b_out, out);
}